// VQCodebook_48361331753022
// MI455X (gfx1250) — compile-verified
//
#include <hip/hip_runtime.h>
#include <hip/hip_bf16.h>

// Problem constants (from reference): B=16, D=128, H=W=32, K=8192
#define NROWS 16384   // B*H*W
#define KCODES 8192
#define DDIM 128
#define HWSZ 1024     // H*W
#define TILEC 32      // codes staged per iteration
#define NTILES (KCODES / TILEC)       // 256
#define LDSPITCH 132                  // 128 + 4 pad floats (bank-conflict-free)

typedef __attribute__((ext_vector_type(2))) float    v2f;
typedef __attribute__((ext_vector_type(8))) float    v8f;
typedef __attribute__((ext_vector_type(4))) unsigned v4u;
typedef __attribute__((ext_vector_type(8))) int      v8i;
typedef __attribute__((ext_vector_type(4))) int      v4i;

// ---------------------------------------------------------------------------
// Kernel 1: per-code squared norms ||c_k||^2. One wave32 per code.
// ---------------------------------------------------------------------------
__global__ __launch_bounds__(256) void vq_cnorm_kernel(
    const float* __restrict__ codebook, float* __restrict__ cnorm)
{
    const int lane = threadIdx.x & 31;
    const int wave = threadIdx.x >> 5;
    const int k = blockIdx.x * 8 + wave;
    const float* row = codebook + (size_t)k * DDIM;
    float s = 0.0f;
#pragma unroll
    for (int j = 0; j < 4; ++j) { float x = row[lane + 32 * j]; s += x * x; }
#pragma unroll
    for (int m = 16; m >= 1; m >>= 1) s += __shfl_xor(s, m);
    if (lane == 0) cnorm[k] = s;
}

// ---------------------------------------------------------------------------
// TDM descriptor helper: issue an async 2D tile load (32 codes x 128 f32,
// contiguous 16KB) from global codebook into LDS with hardware row padding
// (pad 4 DWORDs after every 128 DWORDs -> 132-float LDS pitch).
// D# per CDNA5 ISA §8.3-8.6; 6-arg (clang-23) builtin form.
// ---------------------------------------------------------------------------
__device__ __forceinline__ void tdm_load_tile(const float* gsrc, unsigned ldsByteAddr)
{
    const unsigned long long ga = (unsigned long long)(uintptr_t)gsrc;
    v4u g0;
    g0.x = 1u;                                     // count=1 (valid), load, no gather
    g0.y = ldsByteAddr;                            // lds_addr (bytes)
    g0.z = (unsigned)(ga & 0xFFFFFFFFu);           // global_addr[31:0]
    g0.w = (unsigned)((ga >> 32) & 0x1FFFFFFu)     // global_addr[56:32]
         | (2u << 30);                             // type = 2 ("image")
    v8i g1;
    g1[0] = (int)((2u << 16)                       // data_size = 4 bytes
               | (1u << 20)                        // pad_enable
               | (6u << 22)                        // pad_interval: 2^(6+1)=128 DWORDs
               | (3u << 25));                      // pad_amount: 3+1 = 4 DWORDs
    g1[1] = (int)(128u << 16);                     // tensor_dim0 = 128 (low16)
    g1[2] = (int)(32u  << 16);                     // tensor_dim1 = 32  (low16)
    g1[3] = (int)(128u << 16);                     // tile_dim0 = 128
    g1[4] = 32;                                    // tile_dim1 = 32 (tile_dim2 = 0)
    g1[5] = 128;                                   // tensor_dim0_stride = 128 (low32)
    g1[6] = 0;
    g1[7] = 0;
    v4i gz4 = {0, 0, 0, 0};                        // 2D: groups 2/3 unused
    v8i gz8 = {0, 0, 0, 0, 0, 0, 0, 0};
    __builtin_amdgcn_tensor_load_to_lds(g0, g1, gz4, gz4, gz8, 0);
}

// ---------------------------------------------------------------------------
// Kernel 2: fused distance-GEMM + argmin using V_WMMA_F32_16X16X4_F32.
// 4 waves/WG, 16 rows/wave (64 rows/WG). Double-buffered LDS tiles staged by
// the Tensor Data Mover (wave 0), overlapped with 64 chained WMMAs (2 chains).
// argmin over k of (||c_k||^2 - 2 z.c_k)  (||z||^2 constant per row).
// ---------------------------------------------------------------------------
__global__ __launch_bounds__(128) void vq_argmin_wmma_kernel(
    const float* __restrict__ z_e,       // (B,D,H,W)
    const float* __restrict__ codebook,  // (K,D)
    const float* __restrict__ cnorm,     // (K,)
    int*   __restrict__ idxW,            // ws: winning index per row
    float* __restrict__ outIdxF,         // d_out: indices as float
    float* __restrict__ countW)          // ws: cluster counts (atomic)
{
    __shared__ float ldsB[2][TILEC * LDSPITCH];   // 2 x 32 codes x 128 (padded)

    const int tid  = threadIdx.x;
    const int lane = tid & 31;
    const int wave = tid >> 5;
    const int half = lane >> 4;          // 0: lanes 0-15, 1: lanes 16-31
    const int mlo  = lane & 15;
    const int rowBase = blockIdx.x * 64 + wave * 16;

    // LDS byte addresses of the two buffers (flat addr low 32 bits == DS addr)
    const unsigned ldsA0 = (unsigned)(uintptr_t)(void*)&ldsB[0][0];
    const unsigned ldsA1 = (unsigned)(uintptr_t)(void*)&ldsB[1][0];

    // ---- Preload A fragments (16 rows x 128 D) in WMMA 16x4 f32 layout:
    // lane<16 holds K={4c,4c+1}, lane>=16 holds K={4c+2,4c+3} for row m=lane&15
    const int n  = rowBase + mlo;        // flat row = b*1024 + hw
    const int b  = n >> 10;
    const int hw = n & (HWSZ - 1);
    const float* zb = z_e + (size_t)b * DDIM * HWSZ + hw;   // stride HWSZ per d
    v2f aF[32];
#pragma unroll
    for (int c = 0; c < 32; ++c) {
        const int d0 = 4 * c + 2 * half;
        aF[c].x = zb[(size_t)d0 * HWSZ];
        aF[c].y = zb[(size_t)(d0 + 1) * HWSZ];
    }

    float minv[8];
    int   mini[8];
#pragma unroll
    for (int v = 0; v < 8; ++v) { minv[v] = 3.402823466e38f; mini[v] = 0; }

    // Prologue: DMA tile 0 into buffer 0
    if (wave == 0) {
        tdm_load_tile(codebook, ldsA0);
        __builtin_amdgcn_s_wait_tensorcnt(0);
    }
    __syncthreads();

    for (int t = 0; t < NTILES; ++t) {
        const int tileBase = t * TILEC;

        // Overlap: DMA next tile into the other buffer while computing this one
        if (wave == 0 && (t + 1) < NTILES) {
            tdm_load_tile(codebook + (size_t)(t + 1) * TILEC * DDIM,
                          ((t + 1) & 1) ? ldsA1 : ldsA0);
        }

        const float* buf = &ldsB[t & 1][0];
        v8f acc0 = {};
        v8f acc1 = {};
#pragma unroll
        for (int c = 0; c < 32; ++c) {
            // B fragments (4x16): lane<16 -> K={4c,4c+1}, lane>=16 -> K={4c+2,4c+3}
            const int o0 = mlo * LDSPITCH + 4 * c + 2 * half;
            const int o1 = (16 + mlo) * LDSPITCH + 4 * c + 2 * half;
            v2f b0; b0.x = buf[o0]; b0.y = buf[o0 + 1];
            v2f b1; b1.x = buf[o1]; b1.y = buf[o1 + 1];
            acc0 = __builtin_amdgcn_wmma_f32_16x16x4_f32(
                false, aF[c], false, b0, (short)0, acc0, false, false);
            acc1 = __builtin_amdgcn_wmma_f32_16x16x4_f32(
                false, aF[c], false, b1, (short)0, acc1, false, false);
        }

        // C/D layout: lane holds column ncol=lane&15, rows m = v + 8*half
        const float cn0 = cnorm[tileBase + mlo];
        const float cn1 = cnorm[tileBase + 16 + mlo];
        const int   kc0 = tileBase + mlo;
        const int   kc1 = tileBase + 16 + mlo;
#pragma unroll
        for (int v = 0; v < 8; ++v) {
            const float d0 = cn0 - 2.0f * acc0[v];
            const float d1 = cn1 - 2.0f * acc1[v];
            if (d0 < minv[v]) { minv[v] = d0; mini[v] = kc0; }  // strict < => lowest k on tie
            if (d1 < minv[v]) { minv[v] = d1; mini[v] = kc1; }
        }

        // Ensure next buffer is resident before anyone reads it next iteration
        if (wave == 0 && (t + 1) < NTILES) __builtin_amdgcn_s_wait_tensorcnt(0);
        __syncthreads();
    }

    // Cross-lane argmin within each 16-lane half; prefer smaller index on tie
#pragma unroll
    for (int v = 0; v < 8; ++v) {
        float f = minv[v];
        int   i = mini[v];
#pragma unroll
        for (int m = 8; m >= 1; m >>= 1) {
            const float of = __shfl_xor(f, m);
            const int   oi = __shfl_xor(i, m);
            if (of < f || (of == f && oi < i)) { f = of; i = oi; }
        }
        if (mlo == 0) {
            const int row = rowBase + v + 8 * half;
            idxW[row]    = i;
            outIdxF[row] = (float)i;
            atomicAdd(&countW[i], 1.0f);
        }
    }
}

// ---------------------------------------------------------------------------
// Kernel 3: fused z_q gather (straight-through: ze + (zq - ze)) + dw scatter.
// One block per (b,d) pair -> coalesced z_e/z_q, codebook row gather via L2.
// ---------------------------------------------------------------------------
__global__ __launch_bounds__(256) void vq_gather_scatter_kernel(
    const float* __restrict__ z_e,
    const float* __restrict__ codebook,
    const int*   __restrict__ idxW,
    float* __restrict__ out_zq,
    float* __restrict__ dwW)
{
    const int p = blockIdx.x;            // b*128 + d
    const int b = p >> 7;
    const int d = p & 127;
    const size_t base = (size_t)p * HWSZ;
#pragma unroll
    for (int it = 0; it < 4; ++it) {
        const int hw  = threadIdx.x + 256 * it;
        const int idx = idxW[b * HWSZ + hw];
        const float ze = z_e[base + hw];
        const float zq = codebook[(size_t)idx * DDIM + d];
        out_zq[base + hw] = ze + (zq - ze);       // matches reference STE arithmetic
        atomicAdd(&dwW[(size_t)idx * DDIM + d], ze);
    }
}

// ---------------------------------------------------------------------------
// Kernel 4: EMA finalize -> new_weight, new_count, new_codebook.
// ---------------------------------------------------------------------------
__global__ __launch_bounds__(256) void vq_finalize_kernel(
    const float* __restrict__ ema_count,
    const float* __restrict__ ema_weight,
    const float* __restrict__ countW,
    const float* __restrict__ dwW,
    float* __restrict__ out_codebook,
    float* __restrict__ out_count,
    float* __restrict__ out_weight)
{
    const int i = blockIdx.x * 256 + threadIdx.x;   // over K*D
    const int k = i >> 7;
    const float DEC = 0.99f;
    const float OMD = (float)(1.0 - 0.99);          // matches python 1.0-DECAY -> f32
    const float nw = ema_weight[i] * DEC + OMD * dwW[i];
    const float nc = ema_count[k] * DEC + OMD * countW[k];
    out_weight[i]   = nw;
    out_codebook[i] = nw / fmaxf(nc, 1.0f);
    if ((i & 127) == 0) out_count[k] = nc;
}

// ---------------------------------------------------------------------------
extern "C" void kernel_launch(void* const* d_in, const int* in_sizes, int n_in,
                              void* d_out, int out_size, void* d_ws, size_t ws_size,
                              hipStream_t stream) {
    const float* z_e        = (const float*)d_in[0];  // (16,128,32,32)
    const float* codebook   = (const float*)d_in[1];  // (8192,128)
    const float* ema_count  = (const float*)d_in[2];  // (8192,)
    const float* ema_weight = (const float*)d_in[3];  // (8192,128)

    // d_out: z_q_st | indices | new_codebook | new_count | new_weight
    float* out      = (float*)d_out;
    float* out_zq   = out;                                   // 2,097,152
    float* out_idx  = out_zq + (size_t)NROWS * DDIM;         //    16,384
    float* out_cb   = out_idx + NROWS;                       // 1,048,576
    float* out_cnt  = out_cb + (size_t)KCODES * DDIM;        //     8,192
    float* out_wt   = out_cnt + KCODES;                      // 1,048,576

    // Workspace layout (floats): idx[N] | counts[K] | cnorm[K] | dw[K*D]
    float* wsf    = (float*)d_ws;
    int*   idxW   = (int*)wsf;                 // 16384 ints
    float* countW = wsf + NROWS;               // 8192
    float* cnormW = countW + KCODES;           // 8192
    float* dwW    = cnormW + KCODES;           // 1,048,576

    // Zero counts (+cnorm, overwritten anyway) + dw accumulators
    (void)hipMemsetAsync((void*)countW, 0,
                   (size_t)(KCODES + KCODES + (size_t)KCODES * DDIM) * sizeof(float),
                   stream);

    vq_cnorm_kernel<<<KCODES / 8, 256, 0, stream>>>(codebook, cnormW);
    vq_argmin_wmma_kernel<<<NROWS / 64, 128, 0, stream>>>(
        z_e, codebook, cnormW, idxW, out_idx, countW);
    vq_gather_scatter_kernel<<<16 * DDIM, 256, 0, stream>>>(
        z_e, codebook, idxW, out_zq, dwW);
    vq_finalize_kernel<<<(KCODES * DDIM) / 256, 256, 0, stream>>>(
        ema_count, ema_weight, countW, dwW, out_cb, out_cnt, out_wt);
}